// MaskPre_76038101008585
// MI455X (gfx1250) — compile-verified
//
#include <hip/hip_runtime.h>
#include <hip/hip_bf16.h>

// ---------------- problem constants ----------------
#define Bn   8
#define Cc   32
#define Hh   512
#define Ww   512
#define HW   (Hh*Ww)           // 262144
#define CQ   8
#define Nwin 4096
#define ED   2048
#define EDH  1024
#define EPSV 1e-6f

typedef __attribute__((ext_vector_type(2))) float v2f;
typedef __attribute__((ext_vector_type(8))) float v8f;

__device__ __forceinline__ float leaky(float x) { return x > 0.f ? x : 0.1f * x; }
__device__ __forceinline__ float sigm(float x)  { return 1.f / (1.f + __expf(-x)); }

// DPP8 selector: lane i reads lane sel_i within its 8-lane group (VALU, no LDS).
#define DPP8_SEL(a,b,c,d,e,f,g,h) \
    ((a)|((b)<<3)|((c)<<6)|((d)<<9)|((e)<<12)|((f)<<15)|((g)<<18)|((h)<<21))
#define DPP8_XOR1 DPP8_SEL(1,0,3,2,5,4,7,6)
#define DPP8_XOR2 DPP8_SEL(2,3,0,1,6,7,4,5)
#define DPP8_XOR4 DPP8_SEL(4,5,6,7,0,1,2,3)

template <int SEL>
__device__ __forceinline__ float dpp8f(float x) {
    return __int_as_float(__builtin_amdgcn_mov_dpp8(__float_as_int(x), SEL));
}
// Butterfly sum across each 8-lane group: every lane ends with the group sum.
__device__ __forceinline__ float sum8_dpp(float x) {
    x += dpp8f<DPP8_XOR1>(x);
    x += dpp8f<DPP8_XOR2>(x);
    x += dpp8f<DPP8_XOR4>(x);
    return x;
}

// ---------------- k0: zero the pooled accumulator ----------------
__global__ void k0_zero(float* pooled_acc) {
    pooled_acc[threadIdx.x] = 0.f;   // 64 threads
}

// ---------------- k1: fused dual conv1x1 (WMMA f32 16x16x4) + LN + leaky ----
// One wave handles tiles of 16 consecutive pixels. A = [16px x 32ch] f32,
// B[32x16]: cols 0..7 = w_in^T, cols 8..15 = w_conv^T. 8 chained
// v_wmma_f32_16x16x4_f32 accumulate the K=32 reduction exactly in f32.
// D layout: VGPR r, lanes0-15 -> pixel r, lanes16-31 -> pixel r+8, N=lane&15.
// Cross-channel reductions (channel-mean / LayerNorm stats) are width-8
// butterflies done with DPP8 lane permutes (single-cycle VALU, no LDS).
__global__ void k1_fused(const float* __restrict__ xin,
                         const float* __restrict__ w_in,  const float* __restrict__ b_in,
                         const float* __restrict__ w_cv,  const float* __restrict__ b_cv,
                         const float* __restrict__ ln_w,  const float* __restrict__ ln_b,
                         float* __restrict__ xln, float* __restrict__ xm,
                         float* __restrict__ pooled_acc) {
    __shared__ float lds_pool[CQ];
    const int tid  = threadIdx.x;
    if (tid < CQ) lds_pool[tid] = 0.f;
    __syncthreads();

    const int lane = tid & 31;
    const int wave = tid >> 5;
    const int m    = lane & 15;         // A row for loads / output column N
    const bool hi  = lane >= 16;
    const int n    = m;                 // output channel column (0..15)

    // Preload B operands for all 8 K-steps (weights are tiny, L2-resident).
    v2f bmat[8];
#pragma unroll
    for (int kk = 0; kk < 8; ++kk) {
        const int k0 = kk * 4 + (hi ? 2 : 0);
        float w0, w1;
        if (n < CQ) { w0 = w_in[n * Cc + k0];        w1 = w_in[n * Cc + k0 + 1]; }
        else        { w0 = w_cv[(n - CQ) * Cc + k0]; w1 = w_cv[(n - CQ) * Cc + k0 + 1]; }
        v2f t; t.x = w0; t.y = w1; bmat[kk] = t;
    }
    const float bias = (n < CQ) ? b_in[n] : b_cv[n - CQ];
    const float lw   = ln_w[(n >= CQ) ? (n - CQ) : 0];
    const float lb   = ln_b[(n >= CQ) ? (n - CQ) : 0];

    float pacc = 0.f;
    // 2048 blocks, 64 tiles/block (8 per wave), 16 px/tile -> all B*HW pixels.
#pragma unroll 1
    for (int i = 0; i < 8; ++i) {
        const long tile = (long)blockIdx.x * 64 + wave * 8 + i;
        const long p0   = tile * 16;                 // flat pixel over B*HW
        const int  b    = (int)(p0 / HW);
        const long pix  = p0 - (long)b * HW;
        const float* base = xin + (size_t)b * Cc * HW + pix + m;

        v8f acc = {};
#pragma unroll
        for (int kk = 0; kk < 8; ++kk) {
            const int ka = kk * 4 + (hi ? 2 : 0);
            v2f a;
            a.x = base[(size_t)ka * HW];
            a.y = base[(size_t)(ka + 1) * HW];
            acc = __builtin_amdgcn_wmma_f32_16x16x4_f32(
                      false, a, false, bmat[kk], (short)0, acc, false, false);
        }

#pragma unroll
        for (int r = 0; r < 8; ++r) {
            const int  pm = r + (hi ? 8 : 0);        // pixel within tile
            const float v = acc[r] + bias;
            const float p = (n < CQ) ? leaky(v) : v; // branchA leaky / branchB raw
            const float s = sum8_dpp(p);
            const float u = s * 0.125f;              // lanes<8: xm ; lanes>=8: LN mean
            const float d = v - u;
            const float q = sum8_dpp((n >= CQ) ? d * d : 0.f);
            if (n >= CQ) {
                const float rstd = rsqrtf(q * 0.125f + EPSV);
                float xo = leaky(lw * (d * rstd) + lb);
                xln[((size_t)b * HW + pix + pm) * CQ + (n - CQ)] = xo; // channels-last
                pacc += xo;
            } else if (n == 0) {
                xm[(size_t)b * HW + pix + pm] = u;
            }
        }
    }
    if (n >= CQ) atomicAdd(&lds_pool[n - CQ], pacc);
    __syncthreads();
    if (tid < CQ) {
        const int b = (int)(((long)blockIdx.x * 64 * 16) / HW);
        atomicAdd(&pooled_acc[b * CQ + tid], lds_pool[tid]);
    }
}

// ---------------- k2: 3x3 spatial-attention conv + sigmoid ----------------
__global__ void k2_sa(const float* __restrict__ xln, const float* __restrict__ w_sa,
                      const float* __restrict__ b_sa, float* __restrict__ sa_out) {
    __shared__ float tile[10][34][CQ];
    __shared__ float wgt[72];
    const int tid = threadIdx.x;            // 256
    const int b   = blockIdx.z;
    const int x0  = blockIdx.x * 32, y0 = blockIdx.y * 8;
    if (tid < 72) wgt[tid] = w_sa[tid];     // [ch][ky][kx]
    for (int idx = tid; idx < 340; idx += 256) {
        const int ty = idx / 34, tx = idx % 34;
        const int gy = y0 + ty - 1, gx = x0 + tx - 1;
        if (gy >= 0 && gy < Hh && gx >= 0 && gx < Ww) {
            const float* src = xln + ((size_t)b * HW + (size_t)gy * Ww + gx) * CQ;
#pragma unroll
            for (int c = 0; c < CQ; ++c) tile[ty][tx][c] = src[c];
        } else {
#pragma unroll
            for (int c = 0; c < CQ; ++c) tile[ty][tx][c] = 0.f;
        }
    }
    __syncthreads();
    const int lx = tid & 31, ly = tid >> 5;
    float s = b_sa[0];
#pragma unroll
    for (int ky = 0; ky < 3; ++ky)
#pragma unroll
        for (int kx = 0; kx < 3; ++kx)
#pragma unroll
            for (int c = 0; c < CQ; ++c)
                s += tile[ly + ky][lx + kx][c] * wgt[c * 9 + ky * 3 + kx];
    sa_out[(size_t)b * HW + (size_t)(y0 + ly) * Ww + (x0 + lx)] = sigm(s);
}

// ---------------- k3: window variance (ddof=1) + exact bottom-k mask -------
__global__ void k3_mask(const float* __restrict__ xm, float* __restrict__ mask) {
    __shared__ float var[Nwin];
    const int b = blockIdx.x, tid = threadIdx.x;   // 1024 threads
#pragma unroll 1
    for (int t = 0; t < 4; ++t) {
        const int nw = tid + t * 1024;
        const int wy = nw >> 6, wx = nw & 63;
        const float* base = xm + (size_t)b * HW + (size_t)wy * 8 * Ww + wx * 8;
        float s = 0.f, ss = 0.f;
#pragma unroll
        for (int i = 0; i < 8; ++i)
#pragma unroll
            for (int j = 0; j < 8; ++j) { const float v = base[i * Ww + j]; s += v; ss += v * v; }
        var[nw] = (ss - s * s * (1.f / 64.f)) * (1.f / 63.f);
    }
    __syncthreads();
#pragma unroll 1
    for (int t = 0; t < 4; ++t) {
        const int nw = tid + t * 1024;
        const float vn = var[nw];
        int cnt = 0;
        for (int j = 0; j < Nwin; ++j) {
            const float vj = var[j];
            cnt += (vj < vn) || (vj == vn && j < nw);   // stable rank, matches top_k ties
        }
        mask[b * Nwin + nw] = (cnt >= Nwin / 2) ? 1.f : 0.f;
    }
}

// ---------------- k4a: hidden vectors h0/h1 (row-sums of w_m1) -------------
__global__ void k4_h(const float* __restrict__ w_m1, const float* __restrict__ b_m1,
                     float* __restrict__ h0, float* __restrict__ h1) {
    const int wave = threadIdx.x >> 5, lane = threadIdx.x & 31;
    const int e = blockIdx.x * 8 + wave;            // 128 blocks x 8 waves = 1024 rows
    const float* row = w_m1 + (size_t)e * ED;
    float s = 0.f;
    for (int d = lane; d < ED; d += 32) s += row[d];
#pragma unroll
    for (int sh = 1; sh < 32; sh <<= 1) s += __shfl_xor(s, sh);
    if (lane == 0) {
        const float bb = b_m1[e];
        h1[e] = leaky(s + bb);
        h0[e] = leaky(bb);
    }
}

// ---------------- k4b: output vectors V0/V1 --------------------------------
__global__ void k4_v(const float* __restrict__ w_m2, const float* __restrict__ b_m2,
                     const float* __restrict__ h0, const float* __restrict__ h1,
                     float* __restrict__ V0, float* __restrict__ V1) {
    const int wave = threadIdx.x >> 5, lane = threadIdx.x & 31;
    const int f = blockIdx.x * 8 + wave;            // 256 blocks x 8 waves = 2048 rows
    const float* row = w_m2 + (size_t)f * EDH;
    float s0 = 0.f, s1 = 0.f;
    for (int e = lane; e < EDH; e += 32) { const float w = row[e]; s0 += w * h0[e]; s1 += w * h1[e]; }
#pragma unroll
    for (int sh = 1; sh < 32; sh <<= 1) { s0 += __shfl_xor(s0, sh); s1 += __shfl_xor(s1, sh); }
    if (lane == 0) { V0[f] = s0 + b_m2[f]; V1[f] = s1 + b_m2[f]; }
}

// ---------------- k5: broadcast-scatter out_mask (268 MB, b128 stores) -----
__global__ void k5_expand(const float* __restrict__ mask, const float* __restrict__ V0,
                          const float* __restrict__ V1, float* __restrict__ out) {
    const int row = blockIdx.x;                     // b*Nwin+n, 32768 blocks
    const float4* V = (mask[row] != 0.f) ? (const float4*)V1 : (const float4*)V0;
    float4* o = (float4*)(out + (size_t)row * ED);
    const int tid = threadIdx.x;                    // 256 threads, 2 x float4 each
    o[tid]       = V[tid];
    o[tid + 256] = V[tid + 256];
}

// ---------------- k6: channel-attention head -------------------------------
__global__ void k6_ca(const float* __restrict__ pooled_acc, const float* __restrict__ w_ca,
                      const float* __restrict__ b_ca, float* __restrict__ ca_out) {
    const int tid = threadIdx.x;                    // 256 = 8 batches x 32 channels
    const int b = tid >> 5, c = tid & 31;
    float s = b_ca[c];
#pragma unroll
    for (int ch = 0; ch < CQ; ++ch)
        s += w_ca[c * CQ + ch] * (pooled_acc[b * CQ + ch] * (1.f / (float)HW));
    ca_out[b * Cc + c] = sigm(s);
}

// ---------------- launcher -------------------------------------------------
extern "C" void kernel_launch(void* const* d_in, const int* in_sizes, int n_in,
                              void* d_out, int out_size, void* d_ws, size_t ws_size,
                              hipStream_t stream) {
    const float* input_x = (const float*)d_in[0];
    const float* w_in    = (const float*)d_in[1];
    const float* b_in    = (const float*)d_in[2];
    const float* w_conv  = (const float*)d_in[3];
    const float* b_conv  = (const float*)d_in[4];
    const float* ln_w    = (const float*)d_in[5];
    const float* ln_b    = (const float*)d_in[6];
    const float* w_ca    = (const float*)d_in[7];
    const float* b_ca    = (const float*)d_in[8];
    const float* w_sa    = (const float*)d_in[9];
    const float* b_sa    = (const float*)d_in[10];
    const float* w_m1    = (const float*)d_in[11];
    const float* b_m1    = (const float*)d_in[12];
    const float* w_m2    = (const float*)d_in[13];
    const float* b_m2    = (const float*)d_in[14];

    float* out = (float*)d_out;
    float* out_mask = out;                                   // [8,4096,2048]
    float* out_ca   = out + (size_t)Bn * Nwin * ED;          // [8,32]
    float* out_sa   = out_ca + (size_t)Bn * Cc;              // [8,512,512]

    float* ws = (float*)d_ws;
    float* xln        = ws;                                  // 16,777,216
    float* xm         = xln + (size_t)Bn * HW * CQ;          //  2,097,152
    float* pooled_acc = xm + (size_t)Bn * HW;                //         64
    float* mask       = pooled_acc + Bn * CQ;                //     32,768
    float* h0         = mask + Bn * Nwin;                    //      1,024
    float* h1         = h0 + EDH;                            //      1,024
    float* V0         = h1 + EDH;                            //      2,048
    float* V1         = V0 + ED;                             //      2,048

    hipLaunchKernelGGL(k0_zero, dim3(1), dim3(64), 0, stream, pooled_acc);
    hipLaunchKernelGGL(k1_fused, dim3(2048), dim3(256), 0, stream,
                       input_x, w_in, b_in, w_conv, b_conv, ln_w, ln_b,
                       xln, xm, pooled_acc);
    hipLaunchKernelGGL(k2_sa, dim3(Ww / 32, Hh / 8, Bn), dim3(256), 0, stream,
                       xln, w_sa, b_sa, out_sa);
    hipLaunchKernelGGL(k3_mask, dim3(Bn), dim3(1024), 0, stream, xm, mask);
    hipLaunchKernelGGL(k4_h, dim3(EDH / 8), dim3(256), 0, stream, w_m1, b_m1, h0, h1);
    hipLaunchKernelGGL(k4_v, dim3(ED / 8), dim3(256), 0, stream, w_m2, b_m2, h0, h1, V0, V1);
    hipLaunchKernelGGL(k5_expand, dim3(Bn * Nwin), dim3(256), 0, stream, mask, V0, V1, out_mask);
    hipLaunchKernelGGL(k6_ca, dim3(1), dim3(256), 0, stream, pooled_acc, w_ca, b_ca, out_ca);
}